// Kernel_82626580841034
// MI455X (gfx1250) — compile-verified
//
#include <hip/hip_runtime.h>
#include <math.h>

// ---------------------------------------------------------------------------
// Batched 32x32 RBF Gram build -> Cholesky -> L^{-1} -> Kuu^{-1} = L^{-T} L^{-1}
// One wave32 per batch (D*Q*K = 32768 batches). Final symmetric GEMM runs on
// the CDNA5 matrix pipe via V_WMMA_F32_16X16X4_F32 (f32 precision required by
// the Cholesky), everything else is lane-parallel LDS code.
// ---------------------------------------------------------------------------

typedef float v2f __attribute__((ext_vector_type(2)));
typedef float v8f __attribute__((ext_vector_type(8)));

#define MDIM 32      // inducing points per GP
#define LDA  33      // padded row stride (odd -> conflict-free columns)
#define WPB  4       // wave32s per block

__device__ __forceinline__ float softplus_f(float x) {
    // numerically stable: max(x,0) + log1p(exp(-|x|))
    return fmaxf(x, 0.0f) + log1pf(__expf(-fabsf(x)));
}

__global__ __launch_bounds__(WPB * 32)
void kuu_inverse_kernel(const float* __restrict__ xloc,   // [B,32]
                        const float* __restrict__ noise,  // [B]
                        const float* __restrict__ r_os,   // [B]
                        const float* __restrict__ r_ls,   // [B]
                        float* __restrict__ out,          // [B,32,32]
                        int nbatch)
{
    __shared__ float Ash[WPB][MDIM * LDA];  // A, then L (lower) in place
    __shared__ float Vsh[WPB][MDIM * LDA];  // V = L^{-1} (lower)
    __shared__ float Xsh[WPB][MDIM];

    const int lane = threadIdx.x & 31;
    const int wv   = threadIdx.x >> 5;
    int b = blockIdx.x * WPB + wv;
    if (b >= nbatch) b = nbatch - 1;   // clamp (no early return: keep EXEC full
                                       // for WMMA and barriers uniform)

    float* A = Ash[wv];
    float* V = Vsh[wv];
    float* X = Xsh[wv];

    // ---- load inputs -----------------------------------------------------
    X[lane] = xloc[(size_t)b * MDIM + lane];
    const float nz = noise[b];
    const float os = softplus_f(r_os[b]) + 1e-6f;
    const float ls = softplus_f(r_ls[b]) + 1e-6f;
    const float cc = -0.5f / ls;
    __syncthreads();

    // ---- build A = os*exp(-0.5*(xi-xj)^2/ls) + nz*I ; lane i owns row i --
    const float xi = X[lane];
    #pragma unroll 4
    for (int j = 0; j < MDIM; ++j) {
        float d = xi - X[j];
        float v = os * __expf(cc * d * d);
        if (j == lane) v += nz;
        A[lane * LDA + j] = v;
    }
    __syncthreads();

    // ---- in-place lower Cholesky (right-looking, lane i owns row i) ------
    for (int k = 0; k < MDIM; ++k) {
        float dkk = sqrtf(A[k * LDA + k]);   // broadcast read
        float inv = 1.0f / dkk;
        if (lane == k) A[k * LDA + k] = dkk;
        if (lane > k) {
            float l = A[lane * LDA + k] * inv;
            A[lane * LDA + k] = l;                       // column k -> L
            for (int j = k + 1; j <= lane; ++j)          // trailing update
                A[lane * LDA + j] -= l * A[j * LDA + k]; // A[j][k] broadcast
        }
        __syncthreads();
    }

    // ---- V = L^{-1}: lane j computes column j by forward substitution ----
    {
        const int j = lane;
        for (int i = 0; i < MDIM; ++i) {
            float v;
            if (i < j) {
                v = 0.0f;
            } else if (i == j) {
                v = 1.0f / A[i * LDA + i];
            } else {
                // V[k][j] == 0 for k < j, so uniform-bound reduction is exact
                float s = 0.0f;
                for (int k = 0; k < i; ++k)
                    s += A[i * LDA + k] * V[k * LDA + j]; // A broadcast, V strided
                v = -s * (1.0f / A[i * LDA + i]);
            }
            V[i * LDA + j] = v;
        }
    }
    __syncthreads();

    // ---- out = V^T * V  via V_WMMA_F32_16X16X4_F32 -----------------------
    // A-frag 16x4 layout: lanes 0-15 hold M=lane, K={0,1}; lanes 16-31 hold
    // M=lane-16, K={2,3}. B-frag 4x16 mirrors it with N in place of M.
    // C/D 16x16: VGPR r -> M=r (lanes 0-15) / M=r+8 (lanes 16-31), N=lane&15.
    const int mrow  = lane & 15;
    const int khalf = (lane >> 4) << 1;          // 0 or 2
    const int rofs  = (lane >> 4) << 3;          // 0 or 8 (store row offset)
    float* outb = out + (size_t)b * (MDIM * MDIM);

    #pragma unroll
    for (int ti = 0; ti < 2; ++ti) {
        #pragma unroll
        for (int tj = 0; tj < 2; ++tj) {
            v8f acc = {};
            #pragma unroll
            for (int kb = 0; kb < 8; ++kb) {
                const int k0 = 4 * kb + khalf;
                v2f af, bf;
                // Af[m][k] = V[k][ti*16+m]  (V^T tile)
                af.x = V[(k0 + 0) * LDA + ti * 16 + mrow];
                af.y = V[(k0 + 1) * LDA + ti * 16 + mrow];
                // Bf[k][n] = V[k][tj*16+n]
                bf.x = V[(k0 + 0) * LDA + tj * 16 + mrow];
                bf.y = V[(k0 + 1) * LDA + tj * 16 + mrow];
                acc = __builtin_amdgcn_wmma_f32_16x16x4_f32(
                        /*neg_a=*/false, af, /*neg_b=*/false, bf,
                        /*c_mod=*/(short)0, acc,
                        /*reuse_a=*/false, /*reuse_b=*/false);
            }
            const int col = tj * 16 + mrow;
            #pragma unroll
            for (int r = 0; r < 8; ++r)
                outb[(ti * 16 + rofs + r) * MDIM + col] = acc[r];
        }
    }
}

extern "C" void kernel_launch(void* const* d_in, const int* in_sizes, int n_in,
                              void* d_out, int out_size, void* d_ws, size_t ws_size,
                              hipStream_t stream) {
    const float* xloc  = (const float*)d_in[0];   // induc_loc      [D,Q,K,32]
    const float* noise = (const float*)d_in[1];   // induc_noise    [D,Q,K]
    const float* r_os  = (const float*)d_in[2];   // raw_outputscale[D,Q,K]
    const float* r_ls  = (const float*)d_in[3];   // raw_lengthscale[D,Q,K]
    float* out = (float*)d_out;                   // [D,Q,K,32,32]

    const int nbatch = in_sizes[1];               // D*Q*K = 32768
    const int blocks = (nbatch + WPB - 1) / WPB;  // one wave32 per batch
    kuu_inverse_kernel<<<blocks, WPB * 32, 0, stream>>>(
        xloc, noise, r_os, r_ls, out, nbatch);
}